// AttnRes_75196287418378
// MI455X (gfx1250) — compile-verified
//
#include <hip/hip_runtime.h>

// ---------------------------------------------------------------------------
// AttnRes fusion for MI455X (gfx1250): single-pass streaming, online softmax
// over the layer axis, async global->LDS double buffering, WMMA f32 16x16x4
// for the per-row reductions (sum V^2 and q.w dot), split across all 8 waves.
// ---------------------------------------------------------------------------

typedef __attribute__((ext_vector_type(2))) float v2f;
typedef __attribute__((ext_vector_type(8))) float v8f;

#define N_LAYERS 12
#define BATCH    4
#define TSEQ     2048
#define DIM      768
#define TTILE    16
#define ROWSTRIDE 772                    // 768 + 4 floats pad: 16B-aligned rows, bank spread
#define TILE_F   (TTILE * ROWSTRIDE)     // 12352 floats per tile buffer

// LDS layout (float offsets)
#define QW_OFF     0
#define BUF_OFF    768
#define SCR_OFF    (BUF_OFF + 2 * TILE_F)   // 8 waves x 256 (ss) + 8 x 256 (dot)
#define SCRDOT     (SCR_OFF + 2048)
#define M_OFF      (SCRDOT + 2048)          // 16: running max
#define S_OFF      (M_OFF + 16)             // 16: running denom
#define CORR_OFF   (S_OFF + 16)             // 16: per-iter rescale
#define P_OFF      (CORR_OFF + 16)          // 16: per-iter prob weight
#define SMEM_FLOATS (P_OFF + 16)            // 29632 floats = 118528 bytes

// Issue one 16x768 fp32 tile (48KB) as 12 async b128 loads per wave.
// 3072 16-byte chunks / 256 threads = 12 per thread.
__device__ __forceinline__ void issue_tile_async(const float* __restrict__ gbase,
                                                 unsigned ldsbase, int tid) {
#pragma unroll
  for (int it = 0; it < 12; ++it) {
    const int idx = tid + it * 256;          // 0..3071
    const int t   = idx / 192;               // row 0..15
    const int dc  = idx - t * 192;           // 16B chunk within row
    const float* g = gbase + t * DIM + dc * 4;
    const unsigned l = ldsbase + (unsigned)((t * ROWSTRIDE + dc * 4) * 4);
    asm volatile("global_load_async_to_lds_b128 %0, %1, off"
                 :: "v"(l), "v"(g) : "memory");
  }
}

extern "C" __global__ void __launch_bounds__(256)
attnres_stream_kernel(const float* __restrict__ V, const float* __restrict__ q,
                      const float* __restrict__ w, float* __restrict__ out) {
  extern __shared__ float smem[];
  const int tid  = threadIdx.x;
  const int lane = tid & 31;
  const int wv   = tid >> 5;                 // wave index 0..7
  const int bid  = blockIdx.x;
  const int b    = bid >> 7;                 // / (TSEQ/TTILE) = /128
  const int t0   = (bid & 127) << 4;         // * TTILE

  // qw = q * w into LDS; init online-softmax state
#pragma unroll
  for (int k = 0; k < 3; ++k) {
    const int i = tid + k * 256;
    smem[QW_OFF + i] = q[i] * w[i];
  }
  if (tid < 16) {
    smem[M_OFF + tid] = -__builtin_inff();
    smem[S_OFF + tid] = 0.0f;
  }

  const unsigned ldsbuf0 = (unsigned)(unsigned long long)(smem + BUF_OFF);
  const size_t layerStride = (size_t)BATCH * TSEQ * DIM;
  const size_t baseBT = ((size_t)b * TSEQ + t0) * DIM;

  // prefetch layer 0 into buffer 0
  issue_tile_async(V + baseBT, ldsbuf0, tid);

  float4 h4[12];
#pragma unroll
  for (int k = 0; k < 12; ++k) { h4[k].x = 0.f; h4[k].y = 0.f; h4[k].z = 0.f; h4[k].w = 0.f; }

  // WMMA operand setup (documented 16x4 fp32 A layout; B = all-ones is
  // layout-independent: D rows become exact row sums in every column)
  const int rowA = lane & 15;
  const int koff = (lane < 16) ? 0 : 2;
  v2f one; one.x = 1.0f; one.y = 1.0f;

  for (int n = 0; n < N_LAYERS; ++n) {
    // prefetch next layer into the other buffer, then wait for current tile.
    // 12 issues/wave/tile; async loads complete in order, so <=12 outstanding
    // means the current tile is fully resident in LDS.
    if (n + 1 < N_LAYERS) {
      issue_tile_async(V + baseBT + (size_t)(n + 1) * layerStride,
                       ldsbuf0 + (unsigned)(((n + 1) & 1) * TILE_F * 4), tid);
      asm volatile("s_wait_asynccnt 12" ::: "memory");
    } else {
      asm volatile("s_wait_asynccnt 0" ::: "memory");
    }
    __syncthreads();

    const float* tile = smem + BUF_OFF + (n & 1) * TILE_F;

    // ---- all 8 waves: partial WMMA reductions over a 24-chunk D-slice ----
    {
      const float* arow = tile + rowA * ROWSTRIDE + koff;
      const float* qwp  = smem + QW_OFF + koff;
      v8f ssacc  = {};
      v8f dotacc = {};
      const int c0 = wv * 24;
#pragma unroll 4
      for (int cc = 0; cc < 24; ++cc) {
        const int c = c0 + cc;
        v2f a, aq, asq;
        a.x   = arow[c * 4];
        a.y   = arow[c * 4 + 1];
        aq.x  = a.x * qwp[c * 4];
        aq.y  = a.y * qwp[c * 4 + 1];
        asq.x = a.x * a.x;
        asq.y = a.y * a.y;
        ssacc  = __builtin_amdgcn_wmma_f32_16x16x4_f32(false, asq, false, one,
                                                       (short)0, ssacc, false, false);
        dotacc = __builtin_amdgcn_wmma_f32_16x16x4_f32(false, aq, false, one,
                                                       (short)0, dotacc, false, false);
      }
      // spill partial C/D (VGPR r: lanes0-15 = M=r N=lane; lanes16-31 = M=r+8 N=lane-16)
#pragma unroll
      for (int r = 0; r < 8; ++r) {
        smem[SCR_OFF + wv * 256 + r * 32 + lane] = ssacc[r];
        smem[SCRDOT  + wv * 256 + r * 32 + lane] = dotacc[r];
      }
    }
    __syncthreads();

    // ---- 16 threads: combine partials, logits, online softmax update ----
    if (tid < 16) {
      const int t = tid;
      const int slot = (t < 8) ? (t * 32) : ((t - 8) * 32 + 16);  // column N=0
      float ss = 0.f, dq = 0.f;
#pragma unroll
      for (int p = 0; p < 8; ++p) {
        ss += smem[SCR_OFF + p * 256 + slot];
        dq += smem[SCRDOT  + p * 256 + slot];
      }
      const float logit = dq * rsqrtf(ss * (1.0f / 768.0f) + 1e-6f);
      const float mo = smem[M_OFF + t];
      const float mn = fmaxf(mo, logit);
      const float co = __expf(mo - mn);     // 0 on first layer (mo = -inf)
      const float pp = __expf(logit - mn);
      smem[S_OFF + t]    = smem[S_OFF + t] * co + pp;
      smem[M_OFF + t]    = mn;
      smem[CORR_OFF + t] = co;
      smem[P_OFF + t]    = pp;
    }
    __syncthreads();

    // ---- all waves: rescaled accumulation h = h*corr + p*V (float4) ----
#pragma unroll
    for (int k = 0; k < 12; ++k) {
      const int idx4 = tid + (k << 8);       // 16B chunk id, 0..3071
      const int t    = idx4 / 192;           // row (lanes span <=2 rows: 2-way bcast)
      const float co = smem[CORR_OFF + t];
      const float pp = smem[P_OFF + t];
      const float4 v4 = *(const float4*)(tile + (size_t)idx4 * 4 + t * 4);
      h4[k].x = h4[k].x * co + pp * v4.x;
      h4[k].y = h4[k].y * co + pp * v4.y;
      h4[k].z = h4[k].z * co + pp * v4.z;
      h4[k].w = h4[k].w * co + pp * v4.w;
    }
    __syncthreads();   // protect buffer before it is overwritten by next prefetch
  }

  // normalize and store: global element index of chunk idx4 is exactly 4*idx4,
  // so stores are fully linear (512B contiguous per wave, b128 per lane)
  float4* out4 = (float4*)(out + ((size_t)b * TSEQ + t0) * DIM);
#pragma unroll
  for (int k = 0; k < 12; ++k) {
    const int idx4 = tid + (k << 8);
    const int t    = idx4 / 192;
    const float inv = 1.0f / smem[S_OFF + t];
    float4 r;
    r.x = h4[k].x * inv; r.y = h4[k].y * inv;
    r.z = h4[k].z * inv; r.w = h4[k].w * inv;
    out4[idx4] = r;
  }
}

extern "C" void kernel_launch(void* const* d_in, const int* in_sizes, int n_in,
                              void* d_out, int out_size, void* d_ws, size_t ws_size,
                              hipStream_t stream) {
  (void)in_sizes; (void)n_in; (void)d_ws; (void)ws_size; (void)out_size;
  const float* V = (const float*)d_in[0];   // [12,4,2048,768] f32
  const float* q = (const float*)d_in[1];   // [768] f32
  const float* w = (const float*)d_in[2];   // [768] f32
  float* out = (float*)d_out;               // [4,2048,768] f32

  dim3 grid(BATCH * (TSEQ / TTILE));        // 512 workgroups
  dim3 block(256);                          // 8 wave32
  const size_t shmem = SMEM_FLOATS * sizeof(float);  // ~116 KB
  attnres_stream_kernel<<<grid, block, shmem, stream>>>(V, q, w, out);
}